// iTransformer_25254407700645
// MI455X (gfx1250) — compile-verified
//
#include <hip/hip_runtime.h>
#include <math.h>

typedef __attribute__((ext_vector_type(2))) float v2f;
typedef __attribute__((ext_vector_type(8))) float v8f;

__device__ __forceinline__ v8f wmma4(v2f a, v2f b, v8f c) {
#if defined(__HIP_DEVICE_COMPILE__)
  // D = A(16x4,f32) * B(4x16,f32) + C(16x16,f32)
  return __builtin_amdgcn_wmma_f32_16x16x4_f32(false, a, false, b, (short)0, c,
                                               false, false);
#else
  (void)a; (void)b;
  return c;  // host pass stub; never executed
#endif
}

// ---------------------------------------------------------------------------
// fp32 WMMA GEMM, register-blocked: C[M,N] = A[M,K] @ W[K,N] + bias[N]
// 128 threads = 4 waves. Each wave computes a 64x32 output block as
// 4 (M) x 2 (N) WMMA tiles; per K-step: 4 A-frag b64 + 4 B-frag b32 loads
// feed 8 WMMAs (1.0 VMEM per WMMA). Block tile = 64 x 128.
// Requires M % 64 == 0, N % 128 == 0, K % 4 == 0 (true for all call sites).
// ---------------------------------------------------------------------------
__global__ __launch_bounds__(128) void gemm_wmma_kernel(
    const float* __restrict__ A, const float* __restrict__ W,
    const float* __restrict__ bias, float* __restrict__ C,
    int M, int K, int Nn) {
  const int lane  = threadIdx.x & 31;
  const int wave  = threadIdx.x >> 5;
  const int lhalf = lane >> 4;   // 0: K={0,1}, 1: K={2,3} within a step
  const int lmod  = lane & 15;
  const int tileM = blockIdx.y * 64;               // 4 sub-tiles of 16 rows
  const int tileN = blockIdx.x * 128 + wave * 32;  // 2 sub-tiles of 16 cols

  v8f acc[4][2] = {};
  const float* Arow0 = A + (size_t)(tileM + lmod) * K;
  const size_t Astep = (size_t)16 * K;
  const float* Bcol0 = W + tileN + lmod;

  for (int kk = 0; kk < K; kk += 4) {
    const int k0 = kk + (lhalf << 1);
    v2f a[4], b[2];
#pragma unroll
    for (int r = 0; r < 4; ++r) {
      const float* ap = Arow0 + (size_t)r * Astep + k0;
      a[r].x = ap[0];
      a[r].y = ap[1];
    }
#pragma unroll
    for (int c = 0; c < 2; ++c) {
      const float* bp = Bcol0 + (size_t)k0 * Nn + 16 * c;
      b[c].x = bp[0];
      b[c].y = bp[Nn];
    }
#pragma unroll
    for (int r = 0; r < 4; ++r)
#pragma unroll
      for (int c = 0; c < 2; ++c)
        acc[r][c] = wmma4(a[r], b[c], acc[r][c]);
  }

#pragma unroll
  for (int c = 0; c < 2; ++c) {
    const float bvv = bias ? bias[tileN + 16 * c + lmod] : 0.0f;
#pragma unroll
    for (int r = 0; r < 4; ++r)
#pragma unroll
      for (int j = 0; j < 8; ++j) {
        const int row = tileM + 16 * r + j + (lhalf << 3);  // C: M = j + 8*half
        C[(size_t)row * Nn + tileN + 16 * c + lmod] = acc[r][c][j] + bvv;
      }
  }
}

// ---------------------------------------------------------------------------
// Flash attention, one wave per (b, h, 16-query tile). dh = 64, d = H*dh.
// Q,K,V,O laid out [B, N, H*dh] row-major (token-major).
// ---------------------------------------------------------------------------
__global__ __launch_bounds__(32) void flash_attn_kernel(
    const float* __restrict__ Q, const float* __restrict__ K,
    const float* __restrict__ V, float* __restrict__ O,
    int Nn, int d, int dh) {
  const int qt = blockIdx.x;     // query tile (N/16)
  const int h  = blockIdx.y;     // head
  const int b  = blockIdx.z;     // batch
  const int lane  = threadIdx.x;
  const int lhalf = lane >> 4;
  const int lmod  = lane & 15;

  __shared__ float qs[16][64];     // Q tile (pre-scaled)
  __shared__ float pmat[16][16];   // score / P tile
  __shared__ float mrow[16], lrow[16], frow[16];

  const float scale = rsqrtf((float)dh);
  const size_t base = (size_t)b * Nn * d + (size_t)h * dh;

  for (int idx = lane; idx < 16 * 64; idx += 32) {
    const int r = idx >> 6, c = idx & 63;
    qs[r][c] = Q[base + (size_t)(qt * 16 + r) * d + c] * scale;
  }
  if (lane < 16) { mrow[lane] = -INFINITY; lrow[lane] = 0.0f; }
  __syncthreads();

  v8f oacc[4] = {};   // 16 rows x 64 cols as four 16x16 accumulators

  for (int kt = 0; kt < Nn / 16; ++kt) {
    // --- S = Q * K^T (tile) ---
    v8f s = {};
    const float* Kb = K + base + (size_t)(kt * 16 + lmod) * d;
#pragma unroll
    for (int st = 0; st < 16; ++st) {
      const int k0 = st * 4 + (lhalf << 1);
      v2f a, bb;
      a.x = qs[lmod][k0];
      a.y = qs[lmod][k0 + 1];
      bb.x = Kb[k0];
      bb.y = Kb[k0 + 1];
      s = wmma4(a, bb, s);
    }
#pragma unroll
    for (int j = 0; j < 8; ++j) pmat[j + (lhalf << 3)][lmod] = s[j];
    __syncthreads();

    // --- online softmax row update ---
    if (lane < 16) {
      const int r = lane;
      float mx = mrow[r];
      for (int c = 0; c < 16; ++c) mx = fmaxf(mx, pmat[r][c]);
      const float f = expf(mrow[r] - mx);
      float sum = 0.0f;
      for (int c = 0; c < 16; ++c) {
        const float p = expf(pmat[r][c] - mx);
        pmat[r][c] = p;
        sum += p;
      }
      lrow[r] = lrow[r] * f + sum;
      mrow[r] = mx;
      frow[r] = f;
    }
    __syncthreads();

    // --- O = O*f + P * V(tile) ---
#pragma unroll
    for (int c = 0; c < 4; ++c) {
      v8f oc = oacc[c];
#pragma unroll
      for (int j = 0; j < 8; ++j) oc[j] *= frow[j + (lhalf << 3)];
#pragma unroll
      for (int st = 0; st < 4; ++st) {
        const int k0 = st * 4 + (lhalf << 1);
        v2f a, bb;
        a.x = pmat[lmod][k0];
        a.y = pmat[lmod][k0 + 1];
        const float* Vb = V + base + (size_t)(kt * 16 + k0) * d + c * 16 + lmod;
        bb.x = Vb[0];
        bb.y = Vb[d];
        oc = wmma4(a, bb, oc);
      }
      oacc[c] = oc;
    }
    __syncthreads();
  }

#pragma unroll
  for (int c = 0; c < 4; ++c)
#pragma unroll
    for (int j = 0; j < 8; ++j) {
      const int r = j + (lhalf << 3);
      O[base + (size_t)(qt * 16 + r) * d + c * 16 + lmod] = oacc[c][j] / lrow[r];
    }
}

// ---------------------------------------------------------------------------
// LayerNorm over d=512 (2 elems/thread), optional residual add, in-place safe.
// ---------------------------------------------------------------------------
__global__ __launch_bounds__(256) void ln_kernel(
    const float* __restrict__ x, const float* __restrict__ res,
    const float* __restrict__ w, const float* __restrict__ bparm,
    float* __restrict__ out, int d) {
  const int row = blockIdx.x, t = threadIdx.x;
  const size_t off = (size_t)row * d;
  float v0 = x[off + t], v1 = x[off + t + 256];
  if (res) { v0 += res[off + t]; v1 += res[off + t + 256]; }
  __shared__ float ss[256], sq[256];
  ss[t] = v0 + v1;
  sq[t] = v0 * v0 + v1 * v1;
  __syncthreads();
  for (int st = 128; st > 0; st >>= 1) {
    if (t < st) { ss[t] += ss[t + st]; sq[t] += sq[t + st]; }
    __syncthreads();
  }
  const float mu   = ss[0] / d;
  const float var  = sq[0] / d - mu * mu;
  const float rstd = rsqrtf(var + 1e-5f);
  out[off + t]       = (v0 - mu) * rstd * w[t] + bparm[t];
  out[off + t + 256] = (v1 - mu) * rstd * w[t + 256] + bparm[t + 256];
}

// ---------------------------------------------------------------------------
// RevIN forward: per-(b,n) stats over S, write xn[B,N,S] + mean/std.
// ---------------------------------------------------------------------------
__global__ __launch_bounds__(256) void revin_kernel(
    const float* __restrict__ x, const float* __restrict__ rw,
    const float* __restrict__ rb, float* __restrict__ xn,
    float* __restrict__ meanb, float* __restrict__ stdb,
    int B, int S, int N) {
  const int bn = blockIdx.x;
  const int b = bn / N, n = bn % N;
  const int t = threadIdx.x;
  const float* xb = x + (size_t)b * S * N + n;
  const float x0 = xb[(size_t)t * N];
  const float x1 = xb[(size_t)(t + 256) * N];
  __shared__ float ss[256], sq[256];
  ss[t] = x0 + x1;
  sq[t] = x0 * x0 + x1 * x1;
  __syncthreads();
  for (int st = 128; st > 0; st >>= 1) {
    if (t < st) { ss[t] += ss[t + st]; sq[t] += sq[t + st]; }
    __syncthreads();
  }
  const float mu  = ss[0] / S;
  const float var = sq[0] / S - mu * mu;
  const float sd  = sqrtf(var + 1e-5f);
  const float r   = 1.0f / sd;
  const float w = rw[n], bb = rb[n];
  xn[(size_t)bn * S + t]       = (x0 - mu) * r * w + bb;
  xn[(size_t)bn * S + t + 256] = (x1 - mu) * r * w + bb;
  if (t == 0) { meanb[bn] = mu; stdb[bn] = sd; }
}

// ---------------------------------------------------------------------------
// Gate: logits = h @ gW, softmax over E=8, top-2 -> comb[token, E]
// ---------------------------------------------------------------------------
__global__ __launch_bounds__(64) void gate_top2_kernel(
    const float* __restrict__ h, const float* __restrict__ gW,
    float* __restrict__ comb, int d, int E) {
  const int token = blockIdx.x, t = threadIdx.x;
  __shared__ float hs[512];
  __shared__ float lg[8];
  for (int i = t; i < d; i += 64) hs[i] = h[(size_t)token * d + i];
  __syncthreads();
  if (t < E) {
    float a = 0.0f;
    for (int s = 0; s < d; ++s) a += hs[s] * gW[(size_t)s * E + t];
    lg[t] = a;
  }
  __syncthreads();
  if (t == 0) {
    float mx = lg[0];
    for (int e = 1; e < E; ++e) mx = fmaxf(mx, lg[e]);
    float ex[8], sum = 0.0f;
    for (int e = 0; e < E; ++e) { ex[e] = expf(lg[e] - mx); sum += ex[e]; }
    int i1 = 0; float p1 = ex[0];
    for (int e = 1; e < E; ++e) if (ex[e] > p1) { p1 = ex[e]; i1 = e; }
    int i2 = -1; float p2 = -1.0f;
    for (int e = 0; e < E; ++e)
      if (e != i1 && ex[e] > p2) { p2 = ex[e]; i2 = e; }
    for (int e = 0; e < E; ++e) comb[(size_t)token * E + e] = 0.0f;
    comb[(size_t)token * E + i1] = p1 / sum;
    comb[(size_t)token * E + i2] = p2 / sum;
  }
}

// ---------------------------------------------------------------------------
// MoE FFN (f=8): out[token] = sum_e comb_e * (gelu(h@W1_e + b1_e) @ W2_e + b2_e)
// ---------------------------------------------------------------------------
__global__ __launch_bounds__(64) void moe_ffn_kernel(
    const float* __restrict__ h, const float* __restrict__ comb,
    const float* __restrict__ W1, const float* __restrict__ b1,
    const float* __restrict__ W2, const float* __restrict__ b2,
    float* __restrict__ out, int d, int E, int F) {
  const int token = blockIdx.x, t = threadIdx.x;
  __shared__ float hs[512];
  __shared__ float part[64];
  __shared__ float mid[8];
  __shared__ float cw[8];
  for (int i = t; i < d; i += 64) hs[i] = h[(size_t)token * d + i];
  if (t < E) cw[t] = comb[(size_t)token * E + t];
  __syncthreads();
  float acc[8];
  for (int j = 0; j < 8; ++j) acc[j] = 0.0f;
  for (int e = 0; e < E; ++e) {
    const float we = cw[e];          // uniform across block
    if (we == 0.0f) continue;
    const int col = t >> 3, l8 = t & 7;
    float p = 0.0f;
    const float* W1e = W1 + (size_t)e * d * F;
    for (int s = l8; s < d; s += 8) p += hs[s] * W1e[(size_t)s * F + col];
    part[t] = p;
    __syncthreads();
    if (t < F) {
      float m = b1[(size_t)e * F + t];
      for (int i = 0; i < 8; ++i) m += part[t * 8 + i];
      mid[t] = 0.5f * m * (1.0f + erff(m * 0.70710678118654752f));  // exact gelu
    }
    __syncthreads();
    const float* W2e = W2 + (size_t)e * F * d;
    const float* b2e = b2 + (size_t)e * d;
    for (int j = 0; j < 8; ++j) {
      const int c = t + 64 * j;
      float o = b2e[c];
      for (int ff = 0; ff < F; ++ff) o += mid[ff] * W2e[(size_t)ff * d + c];
      acc[j] += we * o;
    }
    __syncthreads();
  }
  for (int j = 0; j < 8; ++j) out[(size_t)token * d + t + 64 * j] = acc[j];
}

// ---------------------------------------------------------------------------
// Head combine: out += comb[:,e] * tmp   (first pass overwrites)
// ---------------------------------------------------------------------------
__global__ void head_acc_kernel(const float* __restrict__ tmp,
                                const float* __restrict__ comb,
                                float* __restrict__ out,
                                int e, int E, int d, size_t n, int first) {
  const size_t i = (size_t)blockIdx.x * blockDim.x + threadIdx.x;
  if (i >= n) return;
  const size_t token = i / d;
  const float v = comb[token * E + e] * tmp[i];
  out[i] = first ? v : (out[i] + v);
}

// ---------------------------------------------------------------------------
// RevIN denorm: final[b,n,j] = (acc - rb[n]) / (rw[n]+1e-10) * std[b,n] + mean
// ---------------------------------------------------------------------------
__global__ void denorm_kernel(const float* __restrict__ acc,
                              const float* __restrict__ rw,
                              const float* __restrict__ rb,
                              const float* __restrict__ meanb,
                              const float* __restrict__ stdb,
                              float* __restrict__ out, int N, int d, size_t n) {
  const size_t i = (size_t)blockIdx.x * blockDim.x + threadIdx.x;
  if (i >= n) return;
  const size_t bn = i / d;
  const int nn = (int)(bn % N);
  out[i] = (acc[i] - rb[nn]) / (rw[nn] + 1e-10f) * stdb[bn] + meanb[bn];
}

// ---------------------------------------------------------------------------
static void launch_gemm(const float* A, const float* W, const float* bias,
                        float* C, int M, int K, int Nn, hipStream_t s) {
  dim3 grid(Nn / 128, M / 64);
  gemm_wmma_kernel<<<grid, 128, 0, s>>>(A, W, bias, C, M, K, Nn);
}

extern "C" void kernel_launch(void* const* d_in, const int* in_sizes, int n_in,
                              void* d_out, int out_size, void* d_ws,
                              size_t ws_size, hipStream_t stream) {
  (void)in_sizes; (void)n_in; (void)out_size; (void)ws_size;
  const int B = 8, S = 512, N = 1024, d = 512, dh = 64, E = 8, F = 8, L = 2;

  const float* x          = (const float*)d_in[0];
  const float* revin_w    = (const float*)d_in[1];
  const float* revin_b    = (const float*)d_in[2];
  const float* emb_W      = (const float*)d_in[3];
  const float* emb_b      = (const float*)d_in[4];
  const float* Wq         = (const float*)d_in[5];
  const float* bq         = (const float*)d_in[6];
  const float* Wk         = (const float*)d_in[7];
  const float* bk         = (const float*)d_in[8];
  const float* Wv         = (const float*)d_in[9];
  const float* bv         = (const float*)d_in[10];
  const float* Wo         = (const float*)d_in[11];
  const float* bo         = (const float*)d_in[12];
  const float* ln1_w      = (const float*)d_in[13];
  const float* ln1_b      = (const float*)d_in[14];
  const float* ln2_w      = (const float*)d_in[15];
  const float* ln2_b      = (const float*)d_in[16];
  const float* gate_W     = (const float*)d_in[17];
  const float* eW1        = (const float*)d_in[18];
  const float* eb1        = (const float*)d_in[19];
  const float* eW2        = (const float*)d_in[20];
  const float* eb2        = (const float*)d_in[21];
  const float* enc_norm_w = (const float*)d_in[22];
  const float* enc_norm_b = (const float*)d_in[23];
  const float* head_gate  = (const float*)d_in[24];
  const float* head_W     = (const float*)d_in[25];
  const float* head_b     = (const float*)d_in[26];
  float* out = (float*)d_out;

  const size_t MN = (size_t)B * N;      // 8192 tokens
  const size_t TD = MN * (size_t)d;     // 4.19M floats per activation buffer

  float* ws    = (float*)d_ws;
  float* meanb = ws;
  float* stdb  = meanb + MN;
  float* comb  = stdb + MN;
  float* bufA  = comb + MN * E;         // xn / GEMM scratch
  float* bufH  = bufA + TD;             // residual stream h
  float* bufQ  = bufH + TD;
  float* bufK  = bufQ + TD;
  float* bufV  = bufK + TD;
  float* bufT  = bufV + TD;             // attention out / ffn out / head acc

  // RevIN + inverted embedding: h = xn @ emb_W + emb_b
  revin_kernel<<<B * N, 256, 0, stream>>>(x, revin_w, revin_b, bufA, meanb,
                                          stdb, B, S, N);
  launch_gemm(bufA, emb_W, emb_b, bufH, (int)MN, S, d, stream);

  for (int l = 0; l < L; ++l) {
    const size_t wo  = (size_t)l * d * d;
    const size_t bo2 = (size_t)l * d;
    launch_gemm(bufH, Wq + wo, bq + bo2, bufQ, (int)MN, d, d, stream);
    launch_gemm(bufH, Wk + wo, bk + bo2, bufK, (int)MN, d, d, stream);
    launch_gemm(bufH, Wv + wo, bv + bo2, bufV, (int)MN, d, d, stream);
    flash_attn_kernel<<<dim3(N / 16, d / dh, B), 32, 0, stream>>>(
        bufQ, bufK, bufV, bufT, N, d, dh);
    launch_gemm(bufT, Wo + wo, bo + bo2, bufA, (int)MN, d, d, stream);
    ln_kernel<<<(int)MN, 256, 0, stream>>>(bufH, bufA, ln1_w + bo2,
                                           ln1_b + bo2, bufH, d);
    gate_top2_kernel<<<(int)MN, 64, 0, stream>>>(
        bufH, gate_W + (size_t)l * d * E, comb, d, E);
    moe_ffn_kernel<<<(int)MN, 64, 0, stream>>>(
        bufH, comb, eW1 + (size_t)l * E * d * F, eb1 + (size_t)l * E * F,
        eW2 + (size_t)l * E * F * d, eb2 + (size_t)l * E * d, bufT, d, E, F);
    ln_kernel<<<(int)MN, 256, 0, stream>>>(bufH, bufT, ln2_w + bo2,
                                           ln2_b + bo2, bufH, d);
  }

  // Final encoder norm
  ln_kernel<<<(int)MN, 256, 0, stream>>>(bufH, nullptr, enc_norm_w, enc_norm_b,
                                         bufH, d);

  // MoE projection head: 8 dense expert GEMMs + weighted combine
  gate_top2_kernel<<<(int)MN, 64, 0, stream>>>(bufH, head_gate, comb, d, E);
  const int nblk = (int)((TD + 255) / 256);
  for (int e = 0; e < E; ++e) {
    launch_gemm(bufH, head_W + (size_t)e * d * d, head_b + (size_t)e * d, bufA,
                (int)MN, d, d, stream);
    head_acc_kernel<<<nblk, 256, 0, stream>>>(bufA, comb, bufT, e, E, d, TD,
                                              e == 0 ? 1 : 0);
  }

  // RevIN denorm -> final [B, N, d]
  denorm_kernel<<<nblk, 256, 0, stream>>>(bufT, revin_w, revin_b, meanb, stdb,
                                          out, N, d, TD);
}